// LinearAttention_34488587387513
// MI455X (gfx1250) — compile-verified
//
#include <hip/hip_runtime.h>
#include <hip/hip_bf16.h>

typedef __attribute__((ext_vector_type(2))) float v2f;
typedef __attribute__((ext_vector_type(8))) float v8f;

#define S_LEN 4096
#define DHEAD 64
#define BH    64
#define NC    8
#define SROWS_PER_CHUNK 512
#define STAGE 64
#define LDS_STRIDE 68
#define SB_ROWS 128

// workspace layout (floats)
#define OFF_KVP 0
#define SZ_KVP  (BH*NC*DHEAD*DHEAD)
#define OFF_KSP (OFF_KVP + SZ_KVP)
#define SZ_KSP  (BH*NC*DHEAD)
#define OFF_KV  (OFF_KSP + SZ_KSP)
#define SZ_KV   (BH*DHEAD*DHEAD)
#define OFF_KS  (OFF_KV + SZ_KV)

__device__ __forceinline__ float elup1(float x) {
    // elu(x)+1 : x>0 -> x+1 ; else exp(x)
    return x > 0.0f ? x + 1.0f : __expf(x);
}

// ---------------- Phase 1: partial kv_agg = k'^T v, partial k_sum ------------
__global__ __launch_bounds__(256) void la_phase1(const float* __restrict__ k,
                                                 const float* __restrict__ v,
                                                 float* __restrict__ ws) {
    __shared__ float ksh[STAGE * LDS_STRIDE];
    __shared__ float vsh[STAGE * LDS_STRIDE];

    const int tid  = threadIdx.x;
    const int lane = tid & 31;
    const int wave = tid >> 5;
    const int bh    = blockIdx.y;
    const int chunk = blockIdx.x;
    const long sBase0 = (long)bh * S_LEN + (long)chunk * SROWS_PER_CHUNK;

    // wave owns two 16x16 output tiles sharing the same d-block
    const int t0   = wave * 2;
    const int iBlk = (t0 >> 2) * 16;      // d block base
    const int j0   = (t0 & 3) * 16;       // e block bases (0 or 32)
    const int j1   = j0 + 16;

    const int m    = lane & 15;
    const int koff = (lane < 16) ? 0 : 2;

    v8f acc0 = {};
    v8f acc1 = {};
    float ksumAcc = 0.0f;
    const int col    = tid & 63;
    const int rowOff = tid >> 6;          // 0..3

    for (int c = 0; c < SROWS_PER_CHUNK / STAGE; ++c) {
        const long gRow0 = sBase0 + c * STAGE;
        // stage 64 rows of k' and v into padded LDS
        #pragma unroll
        for (int r = 0; r < 16; ++r) {
            const int  row  = r * 4 + rowOff;
            const long gidx = (gRow0 + row) * DHEAD + col;
            const float kp  = elup1(k[gidx]);
            ksh[row * LDS_STRIDE + col] = kp;
            ksumAcc += kp;
            vsh[row * LDS_STRIDE + col] = v[gidx];
        }
        __syncthreads();
        // 16 K-steps of 4 (V_WMMA_F32_16X16X4_F32)
        #pragma unroll
        for (int kb = 0; kb < STAGE; kb += 4) {
            v2f a, b0, b1;
            a.x  = ksh[(kb + koff)     * LDS_STRIDE + iBlk + m];
            a.y  = ksh[(kb + koff + 1) * LDS_STRIDE + iBlk + m];
            b0.x = vsh[(kb + koff)     * LDS_STRIDE + j0 + m];
            b0.y = vsh[(kb + koff + 1) * LDS_STRIDE + j0 + m];
            b1.x = vsh[(kb + koff)     * LDS_STRIDE + j1 + m];
            b1.y = vsh[(kb + koff + 1) * LDS_STRIDE + j1 + m];
            acc0 = __builtin_amdgcn_wmma_f32_16x16x4_f32(false, a, false, b0,
                                                         (short)0, acc0, false, false);
            acc1 = __builtin_amdgcn_wmma_f32_16x16x4_f32(false, a, false, b1,
                                                         (short)0, acc1, false, false);
        }
        __syncthreads();
    }

    // write kv partials: D[m][n], m = r + 8*(lane>=16), n = lane%16
    float* kvp = ws + OFF_KVP + (long)(bh * NC + chunk) * (DHEAD * DHEAD);
    const int hi = (lane >> 4) * 8;
    #pragma unroll
    for (int r = 0; r < 8; ++r) {
        const int dm = iBlk + r + hi;
        kvp[dm * DHEAD + j0 + m] = acc0[r];
        kvp[dm * DHEAD + j1 + m] = acc1[r];
    }

    // k_sum partial: 4 threads per column -> LDS reduce
    ksh[tid] = ksumAcc;
    __syncthreads();
    if (tid < 64) {
        const float s = ksh[tid] + ksh[tid + 64] + ksh[tid + 128] + ksh[tid + 192];
        ws[OFF_KSP + (bh * NC + chunk) * DHEAD + tid] = s;
    }
}

// ---------------- Reduce partials --------------------------------------------
__global__ __launch_bounds__(256) void la_reduce(float* __restrict__ ws) {
    const int bh  = blockIdx.x;
    const int tid = threadIdx.x;
    for (int idx = tid; idx < DHEAD * DHEAD; idx += 256) {
        float s = 0.0f;
        #pragma unroll
        for (int c = 0; c < NC; ++c)
            s += ws[OFF_KVP + (long)(bh * NC + c) * (DHEAD * DHEAD) + idx];
        ws[OFF_KV + (long)bh * (DHEAD * DHEAD) + idx] = s;
    }
    if (tid < 64) {
        float s = 0.0f;
        #pragma unroll
        for (int c = 0; c < NC; ++c)
            s += ws[OFF_KSP + (bh * NC + c) * DHEAD + tid];
        ws[OFF_KS + bh * DHEAD + tid] = s;
    }
}

// ---------------- Phase 2: out = (q' kv_agg) / (q'.k_sum + eps) --------------
__global__ __launch_bounds__(256) void la_phase2(const float* __restrict__ q,
                                                 const float* __restrict__ ws,
                                                 float* __restrict__ out) {
    __shared__ float kvsh[DHEAD * LDS_STRIDE];
    __shared__ float qsh[8 * 16 * LDS_STRIDE];
    __shared__ float ksumsh[DHEAD];
    __shared__ float denomsh[SB_ROWS];

    const int tid  = threadIdx.x;
    const int lane = tid & 31;
    const int wave = tid >> 5;
    const int bh   = blockIdx.y;
    const int sBlk = blockIdx.x * SB_ROWS;

    // stage kv_agg (padded) and k_sum
    const float* kv = ws + OFF_KV + (long)bh * (DHEAD * DHEAD);
    for (int idx = tid; idx < DHEAD * DHEAD; idx += 256)
        kvsh[(idx >> 6) * LDS_STRIDE + (idx & 63)] = kv[idx];
    if (tid < 64) ksumsh[tid] = ws[OFF_KS + bh * DHEAD + tid];

    // stage this wave's 16x64 q' tile (coalesced, padded)
    const int  qbase = wave * 16 * LDS_STRIDE;
    const long gq0   = ((long)bh * S_LEN + sBlk + wave * 16) * DHEAD;
    #pragma unroll
    for (int i = 0; i < 32; ++i) {
        const int idx = i * 32 + lane;
        qsh[qbase + (idx >> 6) * LDS_STRIDE + (idx & 63)] = elup1(q[gq0 + idx]);
    }
    __syncthreads();

    // reciprocal denominator per row
    if (lane < 16) {
        float dsum = 0.0f;
        #pragma unroll
        for (int d = 0; d < DHEAD; ++d)
            dsum += qsh[qbase + lane * LDS_STRIDE + d] * ksumsh[d];
        denomsh[wave * 16 + lane] = 1.0f / (dsum + 1e-6f);
    }
    __syncthreads();

    const int m    = lane & 15;
    const int koff = (lane < 16) ? 0 : 2;
    const int hi   = (lane >> 4) * 8;
    const long gout0 = ((long)bh * S_LEN + sBlk + wave * 16) * DHEAD;

    #pragma unroll
    for (int j = 0; j < 4; ++j) {
        v8f acc = {};
        #pragma unroll
        for (int kb = 0; kb < DHEAD; kb += 4) {
            v2f a, b;
            a.x = qsh[qbase + m * LDS_STRIDE + kb + koff];
            a.y = qsh[qbase + m * LDS_STRIDE + kb + koff + 1];
            b.x = kvsh[(kb + koff)     * LDS_STRIDE + j * 16 + m];
            b.y = kvsh[(kb + koff + 1) * LDS_STRIDE + j * 16 + m];
            acc = __builtin_amdgcn_wmma_f32_16x16x4_f32(false, a, false, b,
                                                        (short)0, acc, false, false);
        }
        #pragma unroll
        for (int r = 0; r < 8; ++r) {
            const int mm = r + hi;
            out[gout0 + mm * DHEAD + j * 16 + m] = acc[r] * denomsh[wave * 16 + mm];
        }
    }
}

extern "C" void kernel_launch(void* const* d_in, const int* in_sizes, int n_in,
                              void* d_out, int out_size, void* d_ws, size_t ws_size,
                              hipStream_t stream) {
    const float* q = (const float*)d_in[0];
    const float* k = (const float*)d_in[1];
    const float* v = (const float*)d_in[2];
    float* out = (float*)d_out;
    float* ws  = (float*)d_ws;

    la_phase1<<<dim3(NC, BH), 256, 0, stream>>>(k, v, ws);
    la_reduce<<<dim3(BH), 256, 0, stream>>>(ws);
    la_phase2<<<dim3(S_LEN / SB_ROWS, BH), 256, 0, stream>>>(q, ws, out);
}